// RaggedMoEGatherModule_4148938408159
// MI455X (gfx1250) — compile-verified
//
#include <hip/hip_runtime.h>
#include <hip/hip_bf16.h>
#include <stdint.h>

// MoE combine: out[t] = w0 * moe_output[slot0] + w1 * moe_output[slot1]
//   w = scores / (sum_k scores + 1e-9)
// Pure HBM-bandwidth-bound (~384 MB moved -> ~16.5us floor @ 23.3 TB/s).
// gfx1250 data path: ASYNCcnt-tracked global->LDS b128 gathers, full row-pair
// staged in LDS (8 loads in flight per wave), in-order asynccnt drain,
// non-temporal b128 stores so the write stream doesn't evict the gather
// working set from L2 (the 256MB table mostly fits in the 192MB L2 and ~26%
// of rows are gathered more than once).

#define HIDDEN 4096
#define TOPK   2
#define CHUNK  1024                 // floats per chunk = 256 threads * 4
#define NCHUNK (HIDDEN / CHUNK)     // 4

typedef float v4f __attribute__((ext_vector_type(4)));

// Async copy 16 bytes per lane: global -> LDS (tracked by ASYNCcnt).
__device__ __forceinline__ void async_gather16(uint32_t lds_off, const float* gp) {
    asm volatile("global_load_async_to_lds_b128 %0, %1, off"
                 :: "v"(lds_off), "v"((uint64_t)(uintptr_t)gp)
                 : "memory");
}

__global__ __launch_bounds__(256)
void moe_combine_kernel(const float* __restrict__ moe_output,
                        const float* __restrict__ scores,
                        const long long* __restrict__ mapped_slots,
                        float* __restrict__ out)
{
    // [chunk][k][floats] = 4*2*1024*4B = 32 KB LDS per block.
    // Each buffer is written exactly once by the async engine and read
    // exactly once by its own lane -> no WAR hazard, no barriers needed.
    __shared__ float lds[NCHUNK][TOPK][CHUNK];

    const int tid = threadIdx.x;        // 0..255
    const int t   = blockIdx.x;         // token id

    // Per-token weights (uniform across block; compiler scalarizes these).
    const float s0  = scores[2 * t + 0];
    const float s1  = scores[2 * t + 1];
    const float w0  = s0 / (s0 + s1 + 1e-9f);
    const float w1  = s1 / (s0 + s1 + 1e-9f);

    const long long m0 = mapped_slots[2 * t + 0];
    const long long m1 = mapped_slots[2 * t + 1];

    const float* r0 = moe_output + (size_t)m0 * HIDDEN + tid * 4;
    const float* r1 = moe_output + (size_t)m1 * HIDDEN + tid * 4;
    float*       op = out        + (size_t)t  * HIDDEN + tid * 4;

    // Prologue: stage the entire 2x16KB row pair; 8 async b128 loads in
    // flight per wave (ASYNCcnt max is 63 -- plenty of headroom).
#pragma unroll
    for (int c = 0; c < NCHUNK; ++c) {
        // flat shared pointer: low 32 bits are the wave-relative LDS address
        const uint32_t lc0 = (uint32_t)(uintptr_t)&lds[c][0][tid * 4];
        const uint32_t lc1 = (uint32_t)(uintptr_t)&lds[c][1][tid * 4];
        async_gather16(lc0, r0 + c * CHUNK);
        async_gather16(lc1, r1 + c * CHUNK);
    }

    // Drain in order: async loads complete in issue order, so after
    // s_wait_asynccnt <= 2*(NCHUNK-1-c) chunk c has landed in LDS.
#pragma unroll
    for (int c = 0; c < NCHUNK; ++c) {
        switch (c) {
            case 0:  asm volatile("s_wait_asynccnt 6" ::: "memory"); break;
            case 1:  asm volatile("s_wait_asynccnt 4" ::: "memory"); break;
            case 2:  asm volatile("s_wait_asynccnt 2" ::: "memory"); break;
            default: asm volatile("s_wait_asynccnt 0" ::: "memory"); break;
        }

        const v4f a = *(const v4f*)&lds[c][0][tid * 4];
        const v4f b = *(const v4f*)&lds[c][1][tid * 4];
        v4f r;
        r.x = w0 * a.x + w1 * b.x;
        r.y = w0 * a.y + w1 * b.y;
        r.z = w0 * a.z + w1 * b.z;
        r.w = w0 * a.w + w1 * b.w;
        // Streaming store: write-once data, keep it out of L2 so it doesn't
        // evict gather-reusable moe_output lines.
        __builtin_nontemporal_store(r, (v4f*)(op + c * CHUNK));
    }
}

extern "C" void kernel_launch(void* const* d_in, const int* in_sizes, int n_in,
                              void* d_out, int out_size, void* d_ws, size_t ws_size,
                              hipStream_t stream) {
    const float*     moe    = (const float*)d_in[0];      // [N_SLOTS, HIDDEN] f32
    const float*     scores = (const float*)d_in[1];      // [T, 2] f32
    const long long* slots  = (const long long*)d_in[2];  // [T, 2] i64
    // d_in[3] = expert_counts: unused by the reference computation.
    float* out = (float*)d_out;                           // [T, HIDDEN] f32

    const int n_tokens = in_sizes[1] / TOPK;
    moe_combine_kernel<<<n_tokens, 256, 0, stream>>>(moe, scores, slots, out);
}